// LSEA_28355374088837
// MI455X (gfx1250) — compile-verified
//
#include <hip/hip_runtime.h>
#include <hip/hip_bf16.h>

// ---------------------------------------------------------------------------
// LSEA block for MI455X (gfx1250, wave32).
// Shapes: B=16, N=4096, K=16 neighbors, C_IN=64, OC2=32, OC=64, OUT=128.
// All channel-mixing steps are f32 WMMA (V_WMMA_F32_16X16X4_F32) GEMMs over
// 16-point tiles; softmax pooling is a lane==channel wave32 kernel.
// ---------------------------------------------------------------------------

typedef __attribute__((ext_vector_type(2))) float v2f;
typedef __attribute__((ext_vector_type(8))) float v8f;

#define BATCH 16
#define NPTS  4096
#define NNEI  16
#define BN    (BATCH * NPTS)      // 65536 points
#define LEAK  0.2f

__device__ __forceinline__ float act_none(float v)  { return v; }
__device__ __forceinline__ float act_relu(float v)  { return v > 0.f ? v : 0.f; }
__device__ __forceinline__ float act_lrelu(float v) { return v > 0.f ? v : LEAK * v; }

// ---------------------------------------------------------------------------
// Generic tiled GEMM + folded-BN + activation:
//   out[m, n] = act( (sum_k A[m,k] * W[n,k]) * s[n] + b[n] )
// One wave computes a 16-row tile (all NOUT columns, 16 at a time).
// KDIM must be a multiple of 4; M must be a multiple of 64 (4 waves/block).
// f32 WMMA fragment layouts per CDNA5 ISA 7.12.2:
//   A 16x4:  lanes 0-15 K={0,1} in vgpr{0,1}; lanes 16-31 K={2,3}
//   B 4x16:  mirrored, lane selects N
//   C/D 16x16: vgpr r -> (M=r, N=lane) lo-half, (M=8+r, N=lane-16) hi-half
// ---------------------------------------------------------------------------
template<int KDIM, int NOUT, int ACT>
__global__ __launch_bounds__(128)
void gemm_sb_act(const float* A, const float* __restrict__ W,
                 const float* __restrict__ s, const float* __restrict__ bvec,
                 float* out, int M)
{
    constexpr int KP = KDIM + 1;  // LDS pad: stride coprime with 64 banks
    __shared__ float Wl[NOUT][KP];
    __shared__ float sl[NOUT], bl[NOUT];
    __shared__ float Al[4][16][KP];

    const int t = threadIdx.x;
    for (int i = t; i < NOUT * KDIM; i += 128)
        Wl[i / KDIM][i % KDIM] = W[i];
    for (int i = t; i < NOUT; i += 128) { sl[i] = s[i]; bl[i] = bvec[i]; }

    const int wave = t >> 5;
    const int lane = t & 31;
    const int m0   = (blockIdx.x * 4 + wave) * 16;

    for (int i = lane; i < 16 * KDIM; i += 32) {
        int r = i / KDIM, c = i % KDIM;
        Al[wave][r][c] = A[(m0 + r) * KDIM + c];
    }
    __syncthreads();

    const int  lm = lane & 15;
    const bool hi = lane >= 16;
    const int  k0 = hi ? 2 : 0, k1 = hi ? 3 : 1;
    const int  rbase = hi ? 8 : 0;

    for (int nb = 0; nb < NOUT; nb += 16) {
        v8f acc = {0.f, 0.f, 0.f, 0.f, 0.f, 0.f, 0.f, 0.f};
        #pragma unroll
        for (int k = 0; k < KDIM; k += 4) {
            v2f a, b;
            a.x = Al[wave][lm][k + k0];
            a.y = Al[wave][lm][k + k1];
            b.x = Wl[nb + lm][k + k0];
            b.y = Wl[nb + lm][k + k1];
            acc = __builtin_amdgcn_wmma_f32_16x16x4_f32(
                false, a, false, b, (short)0, acc, false, false);
        }
        const int   col = nb + lm;
        const float sc = sl[col], bb = bl[col];
        #pragma unroll
        for (int r = 0; r < 8; ++r) {
            float v = acc[r] * sc + bb;
            if (ACT == 1) v = act_relu(v);
            if (ACT == 2) v = act_lrelu(v);
            out[(m0 + rbase + r) * NOUT + col] = v;
        }
    }
}

// ---------------------------------------------------------------------------
// feature [B, 64, N] -> ft [B*N, 64]   (coalesced LDS tile transpose)
// ---------------------------------------------------------------------------
__global__ __launch_bounds__(256)
void transpose_feature(const float* __restrict__ feat, float* __restrict__ ft)
{
    __shared__ float tile[64][65];
    const int b = blockIdx.y;
    const int n0 = blockIdx.x * 64;
    for (int i = threadIdx.x; i < 64 * 64; i += 256) {
        int c = i >> 6, n = i & 63;
        tile[c][n] = feat[(b * 64 + c) * NPTS + n0 + n];
    }
    __syncthreads();
    for (int i = threadIdx.x; i < 64 * 64; i += 256) {
        int n = i >> 6, c = i & 63;
        ft[(b * NPTS + n0 + n) * 64 + c] = tile[c][n];
    }
}

// ---------------------------------------------------------------------------
// Geometry build + lse1 GEMM.  One wave per point: 16 neighbor rows x 10
// geometry features (padded to 12) -> 32 channels, lrelu.
// ---------------------------------------------------------------------------
__global__ __launch_bounds__(128)
void geom_lse1(const float* __restrict__ x, const int* __restrict__ nbr,
               const float* __restrict__ W, const float* __restrict__ s,
               const float* __restrict__ bvec, float* __restrict__ xinfo)
{
    __shared__ float Wl[32][13];
    __shared__ float sl[32], bl[32];
    __shared__ float Al[4][16][13];

    const int t = threadIdx.x;
    for (int i = t; i < 32 * 12; i += 128) {
        int o = i / 12, c = i % 12;
        Wl[o][c] = (c < 10) ? W[o * 10 + c] : 0.f;
    }
    for (int i = t; i < 32; i += 128) { sl[i] = s[i]; bl[i] = bvec[i]; }

    const int wave = t >> 5;
    const int lane = t & 31;
    const int p = blockIdx.x * 4 + wave;         // global point 0..BN-1
    const int b = p >> 12;
    const int n = p & (NPTS - 1);
    const float* xb = x + b * 3 * NPTS;

    const float cx = xb[n], cy = xb[NPTS + n], cz = xb[2 * NPTS + n];
    if (lane < 16) {
        const int id = nbr[p * NNEI + lane];
        const float kx = xb[id], ky = xb[NPTS + id], kz = xb[2 * NPTS + id];
        const float dx = kx - cx, dy = ky - cy, dz = kz - cz;
        const float dist = sqrtf(dx * dx + dy * dy + dz * dz + 1e-12f);
        float* row = Al[wave][lane];
        row[0] = cx; row[1] = cy; row[2] = cz;
        row[3] = kx; row[4] = ky; row[5] = kz;
        row[6] = dx; row[7] = dy; row[8] = dz;
        row[9] = dist; row[10] = 0.f; row[11] = 0.f;
    }
    __syncthreads();

    const int  lm = lane & 15;
    const bool hi = lane >= 16;
    const int  k0 = hi ? 2 : 0, k1 = hi ? 3 : 1;
    const int  rbase = hi ? 8 : 0;

    for (int nb2 = 0; nb2 < 32; nb2 += 16) {
        v8f acc = {0.f, 0.f, 0.f, 0.f, 0.f, 0.f, 0.f, 0.f};
        #pragma unroll
        for (int k = 0; k < 12; k += 4) {
            v2f a, bfr;
            a.x   = Al[wave][lm][k + k0];
            a.y   = Al[wave][lm][k + k1];
            bfr.x = Wl[nb2 + lm][k + k0];
            bfr.y = Wl[nb2 + lm][k + k1];
            acc = __builtin_amdgcn_wmma_f32_16x16x4_f32(
                false, a, false, bfr, (short)0, acc, false, false);
        }
        const int   col = nb2 + lm;
        const float sc = sl[col], bb = bl[col];
        #pragma unroll
        for (int r = 0; r < 8; ++r) {
            float v = act_lrelu(acc[r] * sc + bb);
            xinfo[(p * NNEI + rbase + r) * 32 + col] = v;
        }
    }
}

// ---------------------------------------------------------------------------
// Attentive pooling core (both pools: C=32 channels == wave32 lanes).
//   W = xinfo * (K[nbr] - q);  a = softmax_k(W);  out = sum_k a * V[nbr]
// One wave per point; lane = channel.  Two passes keep W in registers.
// ---------------------------------------------------------------------------
__global__ __launch_bounds__(256)
void attn_pool(const float* __restrict__ xinfo, const float* __restrict__ q,
               const float* __restrict__ kf, const float* __restrict__ vf,
               const int* __restrict__ nbr, float* __restrict__ out)
{
    const int wave = threadIdx.x >> 5;
    const int lane = threadIdx.x & 31;
    const int p = blockIdx.x * 8 + wave;
    const int pbase = p & ~(NPTS - 1);           // b * NPTS

    const float qc = q[p * 32 + lane];
    int   ids[NNEI];
    float w[NNEI];
    float mx = -3.4e38f;
    #pragma unroll
    for (int j = 0; j < NNEI; ++j) {
        ids[j] = pbase + nbr[p * NNEI + j];
        const float kn = kf[ids[j] * 32 + lane];
        const float wv = xinfo[(p * NNEI + j) * 32 + lane] * (kn - qc);
        w[j] = wv;
        mx = fmaxf(mx, wv);
    }
    float sum = 0.f, acc = 0.f;
    #pragma unroll
    for (int j = 0; j < NNEI; ++j) {
        const float e = __expf(w[j] - mx);       // v_exp_f32
        sum += e;
        acc += e * vf[ids[j] * 32 + lane];
    }
    out[p * 32 + lane] = acc / sum;
}

// ---------------------------------------------------------------------------
// out[b, c, n] = lrelu(tA[p, c] + tB[p, c]) with coalesced LDS transpose.
// ---------------------------------------------------------------------------
__global__ __launch_bounds__(256)
void add_lrelu_transpose(const float* __restrict__ tA, const float* __restrict__ tB,
                         float* __restrict__ out)
{
    __shared__ float tile[64][129];
    const int b = blockIdx.y;
    const int n0 = blockIdx.x * 64;
    for (int i = threadIdx.x; i < 64 * 128; i += 256) {
        int n = i >> 7, c = i & 127;
        int p = (b << 12) + n0 + n;
        float v = tA[p * 128 + c] + tB[p * 128 + c];
        tile[n][c] = act_lrelu(v);
    }
    __syncthreads();
    for (int i = threadIdx.x; i < 64 * 128; i += 256) {
        int c = i >> 6, n = i & 63;
        out[(b * 128 + c) * NPTS + n0 + n] = tile[n][c];
    }
}

// ---------------------------------------------------------------------------
extern "C" void kernel_launch(void* const* d_in, const int* in_sizes, int n_in,
                              void* d_out, int out_size, void* d_ws, size_t ws_size,
                              hipStream_t stream)
{
    const float* x       = (const float*)d_in[0];
    const float* feature = (const float*)d_in[1];
    const int*   nbr     = (const int*)d_in[2];
    const float* mlp1_w = (const float*)d_in[3];
    const float* mlp1_s = (const float*)d_in[4];
    const float* mlp1_b = (const float*)d_in[5];
    const float* lse1_w = (const float*)d_in[6];
    const float* lse1_s = (const float*)d_in[7];
    const float* lse1_b = (const float*)d_in[8];
    // p1 block (dict order: p1 then p2, lse2 comes after both)
    const float* p1_wq = (const float*)d_in[9];
    const float* p1_sq = (const float*)d_in[10];
    const float* p1_bq = (const float*)d_in[11];
    const float* p1_wk = (const float*)d_in[12];
    const float* p1_sk = (const float*)d_in[13];
    const float* p1_bk = (const float*)d_in[14];
    const float* p1_wv = (const float*)d_in[15];
    const float* p1_sv = (const float*)d_in[16];
    const float* p1_bv = (const float*)d_in[17];
    const float* p1_w2 = (const float*)d_in[18];
    const float* p1_s2 = (const float*)d_in[19];
    const float* p1_b2 = (const float*)d_in[20];
    const float* p2_wq = (const float*)d_in[21];
    const float* p2_sq = (const float*)d_in[22];
    const float* p2_bq = (const float*)d_in[23];
    const float* p2_wk = (const float*)d_in[24];
    const float* p2_sk = (const float*)d_in[25];
    const float* p2_bk = (const float*)d_in[26];
    const float* p2_wv = (const float*)d_in[27];
    const float* p2_sv = (const float*)d_in[28];
    const float* p2_bv = (const float*)d_in[29];
    const float* p2_w2 = (const float*)d_in[30];
    const float* p2_s2 = (const float*)d_in[31];
    const float* p2_b2 = (const float*)d_in[32];
    const float* lse2_w = (const float*)d_in[33];
    const float* lse2_s = (const float*)d_in[34];
    const float* lse2_b = (const float*)d_in[35];
    const float* mlp2_w = (const float*)d_in[36];
    const float* mlp2_s = (const float*)d_in[37];
    const float* mlp2_b = (const float*)d_in[38];
    const float* res_w  = (const float*)d_in[39];
    const float* res_s  = (const float*)d_in[40];
    const float* res_b  = (const float*)d_in[41];

    // workspace layout (floats)
    float* ws = (float*)d_ws;
    float* ft   = ws;                       // [BN, 64]
    float* f1   = ft   + (size_t)BN * 64;   // [BN, 32]
    float* qb   = f1   + (size_t)BN * 32;
    float* kb   = qb   + (size_t)BN * 32;
    float* vb   = kb   + (size_t)BN * 32;
    float* pl   = vb   + (size_t)BN * 32;   // pool output [BN, 32]
    float* f2   = pl   + (size_t)BN * 32;   // [BN, 32]
    float* f3   = f2   + (size_t)BN * 32;   // [BN, 64]
    float* tA   = f3   + (size_t)BN * 64;   // [BN, 128]
    float* tB   = tA   + (size_t)BN * 128;  // [BN, 128]
    float* xi   = tB   + (size_t)BN * 128;  // x_info [BN*16, 32]

    // 1) feature transpose -> ft
    transpose_feature<<<dim3(NPTS / 64, BATCH), 256, 0, stream>>>(feature, ft);
    // 2) f1 = lrelu(cbn(ft, mlp1))
    gemm_sb_act<64, 32, 2><<<BN / 64, 128, 0, stream>>>(ft, mlp1_w, mlp1_s, mlp1_b, f1, BN);
    // 3) geometry + lse1 -> x_info
    geom_lse1<<<BN / 4, 128, 0, stream>>>(x, nbr, lse1_w, lse1_s, lse1_b, xi);
    // 4) pool-1 q/k/v (relu)
    gemm_sb_act<32, 32, 1><<<BN / 64, 128, 0, stream>>>(f1, p1_wq, p1_sq, p1_bq, qb, BN);
    gemm_sb_act<32, 32, 1><<<BN / 64, 128, 0, stream>>>(f1, p1_wk, p1_sk, p1_bk, kb, BN);
    gemm_sb_act<32, 32, 1><<<BN / 64, 128, 0, stream>>>(f1, p1_wv, p1_sv, p1_bv, vb, BN);
    // 5) pool-1 softmax core
    attn_pool<<<BN / 8, 256, 0, stream>>>(xi, qb, kb, vb, nbr, pl);
    // 6) feat2 = lrelu(cbn(pool, p1_w2))
    gemm_sb_act<32, 32, 2><<<BN / 64, 128, 0, stream>>>(pl, p1_w2, p1_s2, p1_b2, f2, BN);
    // 7) x_info2 = lrelu(cbn(x_info, lse2))  (in place, M = BN*16)
    gemm_sb_act<32, 32, 2><<<(BN * NNEI) / 64, 128, 0, stream>>>(xi, lse2_w, lse2_s, lse2_b, xi, BN * NNEI);
    // 8) pool-2 q/k/v from feat2
    gemm_sb_act<32, 32, 1><<<BN / 64, 128, 0, stream>>>(f2, p2_wq, p2_sq, p2_bq, qb, BN);
    gemm_sb_act<32, 32, 1><<<BN / 64, 128, 0, stream>>>(f2, p2_wk, p2_sk, p2_bk, kb, BN);
    gemm_sb_act<32, 32, 1><<<BN / 64, 128, 0, stream>>>(f2, p2_wv, p2_sv, p2_bv, vb, BN);
    // 9) pool-2 softmax core
    attn_pool<<<BN / 8, 256, 0, stream>>>(xi, qb, kb, vb, nbr, pl);
    // 10) feat3 = lrelu(cbn(pool, p2_w2))  (32 -> 64)
    gemm_sb_act<32, 64, 2><<<BN / 64, 128, 0, stream>>>(pl, p2_w2, p2_s2, p2_b2, f3, BN);
    // 11) out-branch and residual-branch GEMMs (64 -> 128, no activation)
    gemm_sb_act<64, 128, 0><<<BN / 64, 128, 0, stream>>>(f3, mlp2_w, mlp2_s, mlp2_b, tA, BN);
    gemm_sb_act<64, 128, 0><<<BN / 64, 128, 0, stream>>>(ft, res_w, res_s, res_b, tB, BN);
    // 12) add + lrelu + transpose to [B, 128, N]
    add_lrelu_transpose<<<dim3(NPTS / 64, BATCH), 256, 0, stream>>>(tA, tB, (float*)d_out);
}